// GCNBot_74371653697680
// MI455X (gfx1250) — compile-verified
//
#include <hip/hip_runtime.h>
#include <hip/hip_bf16.h>
#include <math.h>

typedef __attribute__((ext_vector_type(2))) float v2f;
typedef __attribute__((ext_vector_type(8))) float v8f;

#define HID 64

// ---------------- degree / norm ----------------
__global__ void k_fill1(float* __restrict__ p, int n) {
  int i = blockIdx.x * blockDim.x + threadIdx.x;
  if (i < n) p[i] = 1.0f;  // self-loop contributes 1 to every degree
}

__global__ void k_deg(const int* __restrict__ dst, float* __restrict__ deg, int E) {
  int i = blockIdx.x * blockDim.x + threadIdx.x;
  if (i < E)
    __hip_atomic_fetch_add(&deg[dst[i]], 1.0f, __ATOMIC_RELAXED, __HIP_MEMORY_SCOPE_AGENT);
}

__global__ void k_rsqrt(float* __restrict__ p, int n) {
  int i = blockIdx.x * blockDim.x + threadIdx.x;
  if (i < n) p[i] = rsqrtf(p[i]);  // deg >= 1 always (self-loops)
}

// ---------------- WMMA f32 GEMM: C[M,64] = A[M,K] @ W[K,64] ----------------
// One block = 16 output rows; 4 waves, wave w owns columns [16w, 16w+16).
// V_WMMA_F32_16X16X4_F32: A 16x4 (lane = M, VGPR pair = K {0,1} or {2,3} by
// lane half), B 4x16 (lane = N, same K striping), C/D 16x16 f32 in 8 VGPRs
// (VGPR r -> M = r + 8*(lane>=16), N = lane&15).
__global__ void __launch_bounds__(128)
k_gemm_wmma(const float* __restrict__ A, const float* __restrict__ W,
            float* __restrict__ C, int M, int K) {
  const int lane = threadIdx.x & 31;
  const int wv   = threadIdx.x >> 5;       // n-tile 0..3
  const int m0   = blockIdx.x * 16;
  const int n0   = wv * 16;
  const int r16  = lane & 15;
  const int kb   = (lane >> 4) << 1;       // 0 or 2

  int mrow = m0 + r16;
  if (mrow >= M) mrow = M - 1;             // clamp loads; EXEC must stay all-1s
  const float* arow = A + (size_t)mrow * K;
  const float* wcol = W + n0 + r16;

  v8f acc = {};
  for (int k = 0; k < K; k += 4) {
    v2f a = *(const v2f*)(arow + k + kb);  // K= kb, kb+1 (8B aligned: kb even)
    v2f b;
    b.x = wcol[(k + kb) * HID];
    b.y = wcol[(k + kb + 1) * HID];
    acc = __builtin_amdgcn_wmma_f32_16x16x4_f32(
        false, a, false, b, (short)0, acc, false, false);
  }

  const int cr0 = (lane >> 4) << 3;        // 0 or 8
#pragma unroll
  for (int r = 0; r < 8; ++r) {
    int m = m0 + cr0 + r;
    if (m < M) C[(size_t)m * HID + n0 + r16] = acc[r];
  }
}

// ---------------- self-loop + bias init: out = h*dinv^2 + b ----------------
__global__ void k_selfbias(const float* __restrict__ h, const float* __restrict__ dinv,
                           const float* __restrict__ b, float* __restrict__ out,
                           int total) {
  int i = blockIdx.x * blockDim.x + threadIdx.x;
  if (i < total) {
    int n = i >> 6, f = i & 63;
    float di = dinv[n];
    out[i] = fmaf(h[i], di * di, b[f]);
  }
}

// ---------------- edge scatter: out[dst] += h[src] * dinv[src]*dinv[dst] ----
// Wave per edge: lane l covers features 2l, 2l+1 (coalesced 64-float row).
__global__ void k_edgeagg(const float* __restrict__ h, const int* __restrict__ src,
                          const int* __restrict__ dst, const float* __restrict__ dinv,
                          float* __restrict__ out, int E) {
  const int lane = threadIdx.x & 31;
  const int wid  = (blockIdx.x * blockDim.x + threadIdx.x) >> 5;
  const int nw   = (gridDim.x * blockDim.x) >> 5;
  for (int e = wid; e < E; e += nw) {
    const int s = src[e];
    const int d = dst[e];
    if (e + nw < E)  // pull next gathered row toward L2/L0 (global_prefetch_b8)
      __builtin_prefetch(h + (size_t)src[e + nw] * HID, 0, 1);
    const float w = dinv[s] * dinv[d];
    v2f hv = *(const v2f*)(h + (size_t)s * HID + (lane << 1));
    float* o = out + (size_t)d * HID + (lane << 1);
    __hip_atomic_fetch_add(o,     hv.x * w, __ATOMIC_RELAXED, __HIP_MEMORY_SCOPE_AGENT);
    __hip_atomic_fetch_add(o + 1, hv.y * w, __ATOMIC_RELAXED, __HIP_MEMORY_SCOPE_AGENT);
  }
}

__global__ void k_relu(float* __restrict__ x, int n) {
  int i = blockIdx.x * blockDim.x + threadIdx.x;
  if (i < n) x[i] = fmaxf(x[i], 0.0f);
}

// ---------------- head: logits = h @ Wlin + blin; log_softmax -------------
__global__ void k_head(const float* __restrict__ h, const float* __restrict__ Wl,
                       const float* __restrict__ bl, float* __restrict__ out, int N) {
  int n = blockIdx.x * blockDim.x + threadIdx.x;
  if (n >= N) return;
  const float* r = h + (size_t)n * HID;
  float l0 = bl[0], l1 = bl[1];
#pragma unroll 8
  for (int k = 0; k < HID; ++k) {
    float v = r[k];
    l0 = fmaf(v, Wl[k * 2 + 0], l0);
    l1 = fmaf(v, Wl[k * 2 + 1], l1);
  }
  float m   = fmaxf(l0, l1);
  float lse = m + logf(expf(l0 - m) + expf(l1 - m));
  out[n * 2 + 0] = l0 - lse;
  out[n * 2 + 1] = l1 - lse;
}

// ---------------- driver ----------------
extern "C" void kernel_launch(void* const* d_in, const int* in_sizes, int n_in,
                              void* d_out, int out_size, void* d_ws, size_t ws_size,
                              hipStream_t stream) {
  (void)n_in; (void)out_size; (void)ws_size;
  const float* x  = (const float*)d_in[0];
  const int*   ei = (const int*)d_in[1];     // [2, E] row-major: src then dst
  const float* W1 = (const float*)d_in[2];
  const float* b1 = (const float*)d_in[3];
  const float* W2 = (const float*)d_in[4];
  const float* b2 = (const float*)d_in[5];
  const float* W3 = (const float*)d_in[6];
  const float* b3 = (const float*)d_in[7];
  const float* Wl = (const float*)d_in[8];
  const float* bl = (const float*)d_in[9];

  const int NF = 128;
  const int N  = in_sizes[0] / NF;
  const int E  = in_sizes[1] / 2;
  const int* src  = ei;
  const int* dstv = ei + E;

  float* dinv = (float*)d_ws;
  size_t off  = (((size_t)N * sizeof(float)) + 255) & ~(size_t)255;
  float* bufA = (float*)((char*)d_ws + off);          // pre-aggregation h
  float* bufB = bufA + (size_t)N * HID;               // aggregated / activated h

  const int T   = 256;
  const int gN  = (N + T - 1) / T;
  const int gE  = (E + T - 1) / T;
  const int gM  = (N + 15) / 16;
  const int tot = N * HID;
  const int gT  = (tot + T - 1) / T;
  const int AGG = 2048;  // 2048 blocks * 8 waves, grid-stride over edges

  // symmetric degree normalization (self-loops included)
  k_fill1<<<gN, T, 0, stream>>>(dinv, N);
  k_deg  <<<gE, T, 0, stream>>>(dstv, dinv, E);
  k_rsqrt<<<gN, T, 0, stream>>>(dinv, N);

  // layer 1: x(100k x 128) @ W1(128 x 64)
  k_gemm_wmma<<<gM, 128, 0, stream>>>(x, W1, bufA, N, NF);
  k_selfbias<<<gT, T, 0, stream>>>(bufA, dinv, b1, bufB, tot);
  k_edgeagg <<<AGG, T, 0, stream>>>(bufA, src, dstv, dinv, bufB, E);
  k_relu    <<<gT, T, 0, stream>>>(bufB, tot);

  // layer 2
  k_gemm_wmma<<<gM, 128, 0, stream>>>(bufB, W2, bufA, N, HID);
  k_selfbias<<<gT, T, 0, stream>>>(bufA, dinv, b2, bufB, tot);
  k_edgeagg <<<AGG, T, 0, stream>>>(bufA, src, dstv, dinv, bufB, E);
  k_relu    <<<gT, T, 0, stream>>>(bufB, tot);

  // layer 3
  k_gemm_wmma<<<gM, 128, 0, stream>>>(bufB, W3, bufA, N, HID);
  k_selfbias<<<gT, T, 0, stream>>>(bufA, dinv, b3, bufB, tot);
  k_edgeagg <<<AGG, T, 0, stream>>>(bufA, src, dstv, dinv, bufB, E);
  k_relu    <<<gT, T, 0, stream>>>(bufB, tot);

  // head + log_softmax
  k_head<<<gN, T, 0, stream>>>(bufB, Wl, bl, (float*)d_out, N);
}